// Slice_22471268893230
// MI455X (gfx1250) — compile-verified
//
#include <hip/hip_runtime.h>
#include <hip/hip_bf16.h>

#define B_DIM   8192
#define H_DIM   512
#define CONVW   7
#define T_DIM   (H_DIM - CONVW + 1)   // 506
#define F_DIM   32
#define TP_DIM  (T_DIM / 2)           // 253
#define OUT_ROW (F_DIM * TP_DIM)      // 8096
#define BN_EPS  1e-5

// ---------------------------------------------------------------------------
// Generic global->LDS staging through the gfx1250 async path (ASYNCcnt),
// falling back to plain loads if the builtins are missing.
// Caller must __syncthreads() afterwards before other threads read the LDS.
// ---------------------------------------------------------------------------
__device__ __forceinline__ void async_copy_ints(const int* __restrict__ gsrc,
                                                int* ldst /*LDS*/, int count,
                                                int tid) {
#if __has_builtin(__builtin_amdgcn_global_load_async_to_lds_b32) && __has_builtin(__builtin_amdgcn_s_wait_asynccnt)
    __attribute__((address_space(1))) int* g =
        (__attribute__((address_space(1))) int*)(int*)gsrc;   // drop const, AS1
    __attribute__((address_space(3))) int* l =
        (__attribute__((address_space(3))) int*)ldst;         // generic -> AS3
    for (int i = tid; i < count; i += 256)
        __builtin_amdgcn_global_load_async_to_lds_b32(g + i, l + i, 0, 0);
    __builtin_amdgcn_s_wait_asynccnt(0);
#else
    for (int i = tid; i < count; i += 256) ldst[i] = gsrc[i];
#endif
}

// Build split hash tables: contrib(v, p) = tabLo[p][v & 63] ^ tabHi[p][v >> 6]
__device__ __forceinline__ void build_tables(const int* __restrict__ meta,
                                             int (*tabLo)[64], int (*tabHi)[64],
                                             int tid) {
    for (int e = tid; e < CONVW * 64; e += 256) {
        int p = e >> 6, idx = e & 63;
        int lo = 0, hi = 0;
#pragma unroll
        for (int i = 0; i < 6; ++i) {
            int m = -((idx >> i) & 1);
            lo ^= meta[p * 12 + i]     & m;
            hi ^= meta[p * 12 + 6 + i] & m;
        }
        tabLo[p][idx] = lo;
        tabHi[p][idx] = hi;
    }
}

__device__ __forceinline__ void hash_row(const int* xs, int* hs,
                                         const int (*tabLo)[64],
                                         const int (*tabHi)[64], int tid) {
    for (int t = tid; t < T_DIM; t += 256) {
        int h = 0;
#pragma unroll
        for (int p = 0; p < CONVW; ++p) {
            int v = xs[t + (CONVW - 1) - p];
            h ^= tabLo[p][v & 63] ^ tabHi[p][(v >> 6) & 63];
        }
        hs[t] = h;
    }
}

// ---------------------------------------------------------------------------
__global__ void k_zero_stats(double* stats) {
    int i = threadIdx.x;
    if (i < 64) stats[i] = 0.0;
}

// Phase 1: hash + gather + pool, accumulate per-feature sum / sumsq.
// Grid: B_DIM/4 blocks of 256; each block handles 4 batch rows.
__global__ void k_stats(const int* __restrict__ x, const int* __restrict__ meta,
                        const float* __restrict__ lut, double* __restrict__ stats,
                        int* __restrict__ hstore) {
    __shared__ int   tabLo[CONVW][64];
    __shared__ int   tabHi[CONVW][64];
    __shared__ int   xs[H_DIM];
    __shared__ int   hs[T_DIM + 2];
    __shared__ float redS[F_DIM];
    __shared__ float redQ[F_DIM];

    const int tid = threadIdx.x;
    const int f   = tid & 31;     // lane == feature
    const int ug  = tid >> 5;     // wave id 0..7

    build_tables(meta, tabLo, tabHi, tid);

    float acc = 0.f, acc2 = 0.f;

    for (int r = 0; r < 4; ++r) {
        const int b = blockIdx.x * 4 + r;
        __syncthreads();
        async_copy_ints(x + (size_t)b * H_DIM, xs, H_DIM, tid);
        __syncthreads();
        hash_row(xs, hs, tabLo, tabHi, tid);
        if (hstore) {
            for (int t = tid; t < T_DIM; t += 256)
                hstore[(size_t)b * T_DIM + t] = hs[t];
        }
        __syncthreads();
        for (int u = ug; u < TP_DIM; u += 8) {
            int h0 = hs[2 * u];
            int h1 = hs[2 * u + 1];
            float pv = lut[(size_t)h0 * F_DIM + f] + lut[(size_t)h1 * F_DIM + f];
            acc  += pv;
            acc2 += pv * pv;
        }
    }

    __syncthreads();
    if (tid < F_DIM) { redS[tid] = 0.f; redQ[tid] = 0.f; }
    __syncthreads();
    atomicAdd(&redS[f], acc);    // ds_add_f32
    atomicAdd(&redQ[f], acc2);
    __syncthreads();
    if (tid < F_DIM) {
        atomicAdd(&stats[tid],          (double)redS[tid]);
        atomicAdd(&stats[F_DIM + tid],  (double)redQ[tid]);
    }
}

// Phase 2: fold BN into per-feature scale/shift.
__global__ void k_finalize(const double* __restrict__ stats,
                           const float* __restrict__ gamma,
                           const float* __restrict__ beta,
                           float* __restrict__ scsh) {
    int fi = threadIdx.x;
    if (fi >= F_DIM) return;
    const double N   = (double)B_DIM * (double)TP_DIM;
    double mean = stats[fi] / N;
    double var  = stats[F_DIM + fi] / N - mean * mean;
    double inv  = 1.0 / sqrt(var + BN_EPS);
    double sc   = (double)gamma[fi] * inv;
    scsh[fi]         = (float)sc;
    scsh[F_DIM + fi] = (float)((double)beta[fi] - mean * sc);
}

// Phase 3: gather + pool + affine, flat tile in LDS, b128 coalesced store.
// Grid: B_DIM blocks of 256; one batch row per block.
__global__ void k_emit(const int* __restrict__ x, const int* __restrict__ meta,
                       const float* __restrict__ lut,
                       const float* __restrict__ scsh,
                       const int* __restrict__ hstore,
                       float* __restrict__ out) {
    __shared__ int hs[T_DIM + 2];
    __shared__ __align__(16) float tileF[OUT_ROW];   // flat [f*253 + u]
    __shared__ float s_sc[F_DIM];
    __shared__ float s_sh[F_DIM];

    const int tid = threadIdx.x;
    const int b   = blockIdx.x;
    const int f   = tid & 31;
    const int ug  = tid >> 5;

    if (tid < F_DIM) { s_sc[tid] = scsh[tid]; s_sh[tid] = scsh[F_DIM + tid]; }

    if (hstore) {
        async_copy_ints(hstore + (size_t)b * T_DIM, hs, T_DIM, tid);
        __syncthreads();
    } else {
        __shared__ int tabLo[CONVW][64];
        __shared__ int tabHi[CONVW][64];
        __shared__ int xs[H_DIM];
        build_tables(meta, tabLo, tabHi, tid);
        async_copy_ints(x + (size_t)b * H_DIM, xs, H_DIM, tid);
        __syncthreads();
        hash_row(xs, hs, tabLo, tabHi, tid);
        __syncthreads();
    }

    const float sc = s_sc[f];
    const float sh = s_sh[f];
    // lane stride through LDS is 253 dwords; 253 % 64 = 61 (coprime with 64)
    // -> all 32 lanes hit distinct banks.
    for (int u = ug; u < TP_DIM; u += 8) {
        int h0 = hs[2 * u];
        int h1 = hs[2 * u + 1];
        float pv = lut[(size_t)h0 * F_DIM + f] + lut[(size_t)h1 * F_DIM + f];
        tileF[f * TP_DIM + u] = pv * sc + sh;
    }
    __syncthreads();

    // coalesced 128-bit writeback: OUT_ROW = 8096 = 4 * 2024
    float4* orow4 = (float4*)(out + (size_t)b * OUT_ROW);
    for (int i = tid; i < OUT_ROW / 4; i += 256) {
        float4 v = *(const float4*)&tileF[i * 4];
        orow4[i] = v;
    }
}

// ---------------------------------------------------------------------------
extern "C" void kernel_launch(void* const* d_in, const int* in_sizes, int n_in,
                              void* d_out, int out_size, void* d_ws, size_t ws_size,
                              hipStream_t stream) {
    const int*   x     = (const int*)d_in[0];
    const int*   meta  = (const int*)d_in[1];
    const float* lut   = (const float*)d_in[2];
    const float* gamma = (const float*)d_in[3];
    const float* beta  = (const float*)d_in[4];
    float*       out   = (float*)d_out;

    // workspace layout: [0,512) stats f64[64]; [512,768) scale/shift f32[64];
    // [1024, 1024 + B*T*4) cached hashes (if it fits)
    double* stats = (double*)d_ws;
    float*  scsh  = (float*)((char*)d_ws + 512);
    size_t  hbytes = (size_t)B_DIM * T_DIM * sizeof(int);
    int*    hstore = (ws_size >= 1024 + hbytes) ? (int*)((char*)d_ws + 1024) : nullptr;

    k_zero_stats<<<1, 64, 0, stream>>>(stats);
    k_stats<<<B_DIM / 4, 256, 0, stream>>>(x, meta, lut, stats, hstore);
    k_finalize<<<1, 64, 0, stream>>>(stats, gamma, beta, scsh);
    k_emit<<<B_DIM, 256, 0, stream>>>(x, meta, lut, scsh, hstore, out);
}